// PairSimilarity_59691455479997
// MI455X (gfx1250) — compile-verified
//
#include <hip/hip_runtime.h>

// PairSimilarity: out = A^T * G * B with scatter matrices A,B (3 nnz/row) and
// G[p1,p2] = (1/l^2 - r^2/l^4) * exp(-r^2/(2 l^2)), r = first_d[p1]-second_d[p2].
//
// Per 16-row p1 tile: stage 1 accumulates T2[p1_local, 3*b+beta] in LDS via
// ds_add_f32; stage 2 is a dense compacted 48x16 x 16x768 GEMM done with
// v_wmma_f32_16x16x4_f32, with results scattered to the L2-resident output
// via global f32 atomics. A-fragments and output-row targets are hoisted into
// registers (chunk/coltile-invariant) so the inner loop is just
// ds_load_2addr + v_wmma + atomics.

typedef __attribute__((ext_vector_type(2))) float v2f;
typedef __attribute__((ext_vector_type(8))) float v8f;

#define TILE_P1 16
#define COLS_PER_CHUNK 768   // 3*256: an atom's 3 columns never straddle a chunk
#define COL_TILES (COLS_PER_CHUNK / 16)   // 48
#define THREADS 256

__global__ void pair_zero_kernel(float* __restrict__ out, size_t n) {
    size_t i = (size_t)blockIdx.x * blockDim.x + threadIdx.x;
    if (i < n) out[i] = 0.0f;
}

__global__ __launch_bounds__(THREADS)
void pair_similarity_kernel(const float* __restrict__ first_d,
                            const float* __restrict__ first_dr,
                            const int*   __restrict__ first_i,
                            const float* __restrict__ second_d,
                            const float* __restrict__ second_dr,
                            const int*   __restrict__ second_i,
                            const float* __restrict__ lengthscale_p,
                            float* __restrict__ out,
                            int P1, int P2, int NCOL) {
    const int tid  = threadIdx.x;
    const int lane = tid & 31;
    const int wave = tid >> 5;                   // 8 waves (wave32)
    const int p1base = blockIdx.x * TILE_P1;

    __shared__ float s_fd[TILE_P1];
    __shared__ int   s_fi[TILE_P1];
    __shared__ float s_fdr[TILE_P1][3];
    __shared__ float s_T2[TILE_P1][COLS_PER_CHUNK];   // 48 KB

    if (tid < TILE_P1) {
        int p1 = p1base + tid;
        if (p1 < P1) {
            s_fd[tid]     = first_d[p1];
            s_fi[tid]     = first_i[p1];
            s_fdr[tid][0] = first_dr[p1 * 3 + 0];
            s_fdr[tid][1] = first_dr[p1 * 3 + 1];
            s_fdr[tid][2] = first_dr[p1 * 3 + 2];
        } else {                                 // dead row: contributes 0
            s_fd[tid] = 0.0f; s_fi[tid] = 0;
            s_fdr[tid][0] = 0.0f; s_fdr[tid][1] = 0.0f; s_fdr[tid][2] = 0.0f;
        }
    }
    const float l      = lengthscale_p[0];
    const float inv_l2 = 1.0f / (l * l);
    const float inv_l4 = inv_l2 * inv_l2;
    __syncthreads();

    // ---- Hoisted, chunk-invariant state (lives in VGPRs) ----
    const int m     = lane & 15;                 // A/D row (M) or B/D col (N)
    const int k0    = (lane < 16) ? 0 : 2;       // K split across half-waves
    const int half8 = (lane < 16) ? 0 : 8;       // D-row offset for this half

    // A fragments: Ahat[r][j] = first_dr[j][r-3j] if 3j <= r < 3j+3 else 0.
    v2f af[3][4];
    // Scatter targets: compact row rloc -> global out row 3*first_i[rloc/3] + rloc%3.
    int grow[3][8];
#pragma unroll
    for (int rt = 0; rt < 3; ++rt) {
        int r = rt * 16 + m;                     // compact output row 0..47
#pragma unroll
        for (int kc = 0; kc < 4; ++kc) {
            int j0  = kc * 4 + k0;
            int d0j = r - 3 * j0;
            int d1j = r - 3 * (j0 + 1);
            v2f a;
            a.x = (d0j >= 0 && d0j < 3) ? s_fdr[j0][d0j]     : 0.0f;
            a.y = (d1j >= 0 && d1j < 3) ? s_fdr[j0 + 1][d1j] : 0.0f;
            af[rt][kc] = a;
        }
#pragma unroll
        for (int v = 0; v < 8; ++v) {
            int rloc  = rt * 16 + v + half8;     // 0..47
            int j     = rloc / 3;
            int alpha = rloc - 3 * j;
            grow[rt][v] = 3 * s_fi[j] + alpha;
        }
    }

    const int nchunks = (NCOL + COLS_PER_CHUNK - 1) / COLS_PER_CHUNK;

    for (int chunk = 0; chunk < nchunks; ++chunk) {
        const int colbase = chunk * COLS_PER_CHUNK;
        const int atom_lo = colbase / 3;
        const int atom_hi = atom_lo + COLS_PER_CHUNK / 3;

        // ---- zero the LDS T2 tile ----
        for (int idx = tid; idx < TILE_P1 * COLS_PER_CHUNK; idx += THREADS)
            (&s_T2[0][0])[idx] = 0.0f;
        __syncthreads();

        // ---- Stage 1: scatter-accumulate T2 in LDS (ds_add_f32) ----
        for (int p2 = tid; p2 < P2; p2 += THREADS) {
            int b = second_i[p2];
            if (b >= atom_lo && b < atom_hi) {
                float sdv = second_d[p2];
                float d0 = second_dr[p2 * 3 + 0];
                float d1 = second_dr[p2 * 3 + 1];
                float d2 = second_dr[p2 * 3 + 2];
                int c = 3 * b - colbase;
#pragma unroll
                for (int j = 0; j < TILE_P1; ++j) {
                    float r  = s_fd[j] - sdv;
                    float r2 = r * r;
                    float kv = __expf(-0.5f * r2 * inv_l2);
                    float gg = (inv_l2 - r2 * inv_l4) * kv;
                    atomicAdd(&s_T2[j][c + 0], gg * d0);
                    atomicAdd(&s_T2[j][c + 1], gg * d1);
                    atomicAdd(&s_T2[j][c + 2], gg * d2);
                }
            }
        }
        __syncthreads();

        // ---- Stage 2: U = Ahat(48x16) * T2(16x768) via WMMA f32 16x16x4,
        //      scatter U rows to out with global f32 atomics. ----
#pragma unroll
        for (int rt = 0; rt < 3; ++rt) {
            for (int ct = wave; ct < COL_TILES; ct += 8) {   // uniform per wave
                int col0 = ct * 16;
                v8f acc = {};
#pragma unroll
                for (int kc = 0; kc < 4; ++kc) {
                    int j0 = kc * 4 + k0;
                    v2f bf;
                    bf.x = s_T2[j0][col0 + m];
                    bf.y = s_T2[j0 + 1][col0 + m];
                    acc = __builtin_amdgcn_wmma_f32_16x16x4_f32(
                            false, af[rt][kc], false, bf, (short)0, acc,
                            false, false);
                }
                int gcol = colbase + col0 + m;
                if (gcol < NCOL) {
#pragma unroll
                    for (int v = 0; v < 8; ++v)
                        atomicAdd(&out[(size_t)grow[rt][v] * NCOL + gcol],
                                  acc[v]);
                }
            }
        }
        __syncthreads();
    }
}

extern "C" void kernel_launch(void* const* d_in, const int* in_sizes, int n_in,
                              void* d_out, int out_size, void* d_ws, size_t ws_size,
                              hipStream_t stream) {
    const float* first_d   = (const float*)d_in[0];
    const float* first_dr  = (const float*)d_in[1];
    const int*   first_i   = (const int*)  d_in[2];
    const float* second_d  = (const float*)d_in[3];
    const float* second_dr = (const float*)d_in[4];
    const int*   second_i  = (const int*)  d_in[5];
    const float* lsc       = (const float*)d_in[6];
    float* out = (float*)d_out;

    const int P1 = in_sizes[0];
    const int P2 = in_sizes[3];
    const int N2 = 1000;               // reference constant
    const int NCOL = 3 * N2;           // out is (3*N1) x (3*N2)

    // Zero the accumulation buffer every call (harness poisons d_out).
    size_t n = (size_t)out_size;
    pair_zero_kernel<<<(unsigned)((n + 1023) / 1024), 1024, 0, stream>>>(out, n);

    int ntiles = (P1 + TILE_P1 - 1) / TILE_P1;   // 250 workgroups
    pair_similarity_kernel<<<ntiles, THREADS, 0, stream>>>(
        first_d, first_dr, first_i, second_d, second_dr, second_i, lsc,
        out, P1, P2, NCOL);
}